// AperiodicityExtractionByTANDEM_53678501266251
// MI455X (gfx1250) — compile-verified
//
#include <hip/hip_runtime.h>

typedef float v2f __attribute__((ext_vector_type(2)));
typedef float v8f __attribute__((ext_vector_type(8)));

#define BB 8
#define TT 320000
#define NN 4000

__constant__ float c_hhp[41] = {
  0.00041447996898231424f,  0.0007812505141729248f, -0.0010917236836275842f,
 -0.001986792567596759f,    0.0020903896961562292f,  0.004094057027284935f,
 -0.0034025808529816698f,  -0.007496154127205602f,   0.004972263339933064f,
  0.012738791249119802f,   -0.006696032689574911f,  -0.020694051570247052f,
  0.008432436565041345f,    0.03307438375870053f,   -0.010018936738799522f,
 -0.05423136140580825f,     0.011293988915051487f,   0.10020081367388213f,
 -0.012120546202484579f,   -0.316300210390957f,      0.5124068258062764f,
 -0.316300210390957f,      -0.012120546202484579f,   0.10020081367388213f,
  0.011293988915051487f,   -0.05423136140580825f,   -0.010018936738799522f,
  0.03307438375870053f,     0.008432436565041345f,  -0.020694051570247052f,
 -0.006696032689574911f,    0.012738791249119802f,   0.004972263339933064f,
 -0.007496154127205602f,   -0.0034025808529816698f,  0.004094057027284935f,
  0.0020903896961562292f,  -0.001986792567596759f,  -0.0010917236836275842f,
  0.0007812505141729248f,   0.00041447996898231424f
};

__constant__ float c_hlp[37] = {
 -0.0006548817007748305f,   7.561994958159384e-05f,  0.0020408456937895227f,
 -0.0007468053532203044f,  -0.004350223568826493f,   0.0025966428382642732f,
  0.007639602282756696f,   -0.006490411890149785f,  -0.011765804538954506f,
  0.013649908479276255f,    0.01636866479016021f,   -0.026075976030529347f,
 -0.020910294856659444f,    0.04826072503231665f,    0.02476784661104811f,
 -0.09617846758336064f,    -0.027359756709866623f,   0.3148805216163004f,
  0.5282734359405503f,
  0.3148805216163004f,     -0.027359756709866623f,  -0.09617846758336064f,
  0.02476784661104811f,     0.04826072503231665f,   -0.020910294856659444f,
 -0.026075976030529347f,    0.01636866479016021f,    0.013649908479276255f,
 -0.011765804538954506f,   -0.006490411890149785f,   0.007639602282756696f,
  0.0025966428382642732f,  -0.004350223568826493f,  -0.0007468053532203044f,
  0.0020408456937895227f,   7.561994958159384e-05f, -0.0006548817007748305f
};

__device__ __forceinline__ int clampi(int v, int lo, int hi) {
  return v < lo ? lo : (v > hi ? hi : v);
}

// One QMF decomposition level: out length = Lin/2, reflect padding.
__global__ void qmf_kernel(const float* __restrict__ in, float* __restrict__ outH,
                           float* __restrict__ outL, int Lin) {
  const int Lout = Lin >> 1;
  long i = (long)blockIdx.x * blockDim.x + threadIdx.x;
  if (i >= (long)BB * Lout) return;
  int o = (int)(i % Lout);
  int b = (int)(i / Lout);
  const float* xr = in + (long)b * Lin;
  const int base = 2 * o;
  float sh = 0.f, sl = 0.f;
#pragma unroll
  for (int t = 0; t < 41; ++t) {
    int ix = base + t - 20;
    ix = ix < 0 ? -ix : (ix >= Lin ? 2 * Lin - 2 - ix : ix);
    sh = fmaf(c_hhp[t], xr[ix], sh);
  }
#pragma unroll
  for (int t = 0; t < 37; ++t) {
    int ix = base + t - 18;
    ix = ix < 0 ? -ix : (ix >= Lin ? 2 * Lin - 2 - ix : ix);
    sl = fmaf(c_hlp[t], xr[ix], sl);
  }
  outH[(long)b * Lout + o] = sh;
  outL[(long)b * Lout + o] = sl;
}

// Per-band TANDEM aperiodicity. One wave32 handles TWO frames via a chained
// V_WMMA_F32_16X16X4_F32 normal-equation build:
//   A rows 0-5  = frame0 weighted H^T,  rows 6-11 = frame1 weighted H^T
//   B cols 0-6  = frame0 [H | X],       cols 7-13 = frame1 [H | X]
// Block-diagonal parts of D give [R|b] for each frame. Rows 12-15 / cols
// 14-15 of D are never read, and the window is zero-padded past seg, so the
// operand gathers need NO predication: the inner loop is branch-free and
// EXEC stays all-ones across every WMMA.
__global__ void band_kernel(const float* __restrict__ xb, int Lb,
                            const float* __restrict__ f0, float* __restrict__ Ab,
                            int band, float tmp_fs, int seg) {
  __shared__ float s_win[256];
  __shared__ float s_tile[8 * 256];

  const int tid  = threadIdx.x;
  const int wave = tid >> 5;
  const int lane = tid & 31;
  const int g    = lane >> 4;   // K sub-group
  const int r    = lane & 15;   // A row / B col / D col

  // Hanning window, zero-padded to the full 256-entry buffer.
  s_win[tid] = (tid < seg)
      ? 0.5f - 0.5f * cosf(6.28318530717958647692f * (float)(tid + 1) / (float)(seg + 1))
      : 0.f;
  __syncthreads();

  int pairIdx = blockIdx.x * 8 + wave;
  const int nPairs = BB * (NN / 2);
  if (pairIdx >= nPairs) pairIdx = nPairs - 1;  // defensive; grid is exact
  const int b  = pairIdx / (NN / 2);
  const int n0 = (pairIdx % (NN / 2)) * 2;
  const float* xr = xb + (long)b * Lb;
  const int T1 = Lb - 1;

  // Per-frame pitch-synchronous parameters (uniform across the wave)
  float f00 = f0[b * NN + n0];     if (f00 <= 32.f) f00 = 150.f;
  float f01 = f0[b * NN + n0 + 1]; if (f01 <= 32.f) f01 = 150.f;
  const float p0 = tmp_fs / f00, p1 = tmp_fs / f01;
  const int t00 = (int)(p0 + 0.5f),        t01 = (int)(p1 + 0.5f);
  const int ib0 = (int)(p0 * 0.5f + 0.5f), ib1 = (int)(p1 * 0.5f + 0.5f);
  const int cp0 = (int)((float)n0 * 0.005f * tmp_fs + 1.5f);
  const int cp1 = (int)((float)(n0 + 1) * 0.005f * tmp_fs + 1.5f);
  const int origin0 = cp0 - ib0, origin1 = cp1 - ib1;

  // A-operand lane mapping: weighted H^T row (rows >= 12 produce unread junk)
  const int afr = (r < 6) ? 0 : 1;
  const int arr = (r < 6) ? r : ((r < 12) ? r - 6 : 0);
  {
  }
  const int aOrg = afr ? origin1 : origin0;
  const int aT0  = afr ? t01 : t00;
  const int aBase = aOrg + ((arr < 3) ? -aT0 : aT0) + (arr % 3) - 1;

  // B-operand lane mapping: [H | X] column (cols >= 14 produce unread junk)
  const int bfr = (r < 7) ? 0 : 1;
  const int bc  = (r < 7) ? r : ((r < 14) ? r - 7 : 0);
  const int bOrg = bfr ? origin1 : origin0;
  const int bT0  = bfr ? t01 : t00;
  const int bBase = (bc == 6) ? bOrg
                              : bOrg + ((bc < 3) ? -bT0 : bT0) + (bc % 3) - 1;

  // Chained f32 WMMA accumulation over K = seg (chunks of 4); branch-free body.
  v8f acc = {0.f, 0.f, 0.f, 0.f, 0.f, 0.f, 0.f, 0.f};
  const int nch = (seg + 3) >> 2;
  for (int kc = 0; kc < nch; ++kc) {
    const int kx = kc * 4 + 2 * g;  // lane group g holds K = 2g, 2g+1
    const int ky = kx + 1;
    const float wx = s_win[kx];     // zero for k >= seg kills padded K-slices
    const float wy = s_win[ky];
    v2f av, bv;
    av.x = wx * xr[clampi(aBase + kx, 0, T1)];
    av.y = wy * xr[clampi(aBase + ky, 0, T1)];
    bv.x = xr[clampi(bBase + kx, 0, T1)];
    bv.y = xr[clampi(bBase + ky, 0, T1)];
    acc = __builtin_amdgcn_wmma_f32_16x16x4_f32(false, av, false, bv,
                                                (short)0, acc, false, false);
  }

  // Spill D (16x16) to LDS: VGPR i, lane n -> row i + 8*(n>=16), col n&15
  float* tp = &s_tile[wave * 256];
#pragma unroll
  for (int i2 = 0; i2 < 8; ++i2)
    tp[(i2 + 8 * g) * 16 + r] = acc[i2];
  __syncthreads();

  // Lanes 0,1: Cholesky solve (R + eps I) a = b for frame = lane
  if (lane < 2) {
    const int f = lane;
    float M[6][7];
    for (int i2 = 0; i2 < 6; ++i2)
      for (int c = 0; c < 7; ++c)
        M[i2][c] = tp[(i2 + 6 * f) * 16 + (c + 7 * f)];
    for (int i2 = 0; i2 < 6; ++i2) M[i2][i2] += 1e-5f;
    float L[6][6];
    for (int j = 0; j < 6; ++j) {
      float d = M[j][j];
      for (int p = 0; p < j; ++p) d -= L[j][p] * L[j][p];
      d = sqrtf(fmaxf(d, 1e-30f));
      L[j][j] = d;
      for (int i2 = j + 1; i2 < 6; ++i2) {
        float s = M[i2][j];
        for (int p = 0; p < j; ++p) s -= L[i2][p] * L[j][p];
        L[i2][j] = s / d;
      }
    }
    float y[6], a[6];
    for (int i2 = 0; i2 < 6; ++i2) {
      float s = M[i2][6];
      for (int p = 0; p < i2; ++p) s -= L[i2][p] * y[p];
      y[i2] = s / L[i2][i2];
    }
    for (int i2 = 5; i2 >= 0; --i2) {
      float s = y[i2];
      for (int p = i2 + 1; p < 6; ++p) s -= L[p][i2] * a[p];
      a[i2] = s / L[i2][i2];
    }
    for (int c = 0; c < 6; ++c) tp[240 + f * 8 + c] = a[c];
  }
  __syncthreads();

  // Residual pass: lanes 0-15 frame0, lanes 16-31 frame1
  const int fr  = g;
  const int org = fr ? origin1 : origin0;
  const int t0f = fr ? t01 : t00;
  const float* av6 = &tp[240 + fr * 8];
  const float a0 = av6[0], a1 = av6[1], a2 = av6[2];
  const float a3 = av6[3], a4 = av6[4], a5 = av6[5];
  float sv = 0.f, sv2 = 0.f, su = 0.f, su2 = 0.f;
  for (int k = r; k < seg; k += 16) {
    const float X = xr[clampi(org + k, 0, T1)];
    const int ba = org - t0f + k - 1;
    const int bb2 = org + t0f + k - 1;
    float pred = a0 * xr[clampi(ba, 0, T1)];
    pred = fmaf(a1, xr[clampi(ba + 1, 0, T1)], pred);
    pred = fmaf(a2, xr[clampi(ba + 2, 0, T1)], pred);
    pred = fmaf(a3, xr[clampi(bb2, 0, T1)], pred);
    pred = fmaf(a4, xr[clampi(bb2 + 1, 0, T1)], pred);
    pred = fmaf(a5, xr[clampi(bb2 + 2, 0, T1)], pred);
    const float wsq = sqrtf(s_win[k]);
    const float v = wsq * (X - pred);
    const float u = wsq * X;
    sv += v; sv2 = fmaf(v, v, sv2); su += u; su2 = fmaf(u, u, su2);
  }
  // Reduce within each 16-lane half (xor masks < 16 never cross halves)
#pragma unroll
  for (int off = 1; off < 16; off <<= 1) {
    sv  += __shfl_xor(sv,  off);
    sv2 += __shfl_xor(sv2, off);
    su  += __shfl_xor(su,  off);
    su2 += __shfl_xor(su2, off);
  }
  if (r == 0) {
    const float ns = (float)seg;
    const float varv = (sv2 - sv * sv / ns) / (ns - 1.f);
    const float varu = (su2 - su * su / ns) / (ns - 1.f);
    const float Aap = sqrtf(fmaxf(varv, 0.f)) / (sqrtf(fmaxf(varu, 0.f)) + 1e-16f);
    const int nf = n0 + fr;
    Ab[((long)b * NN + nf) * 4 + band] = Aap;
  }
}

// Log-domain interpolation of the 5 band values (band3 duplicated at DC)
// onto 513 FFT bins, then exp.
__global__ void interp_kernel(const float* __restrict__ Ab, float* __restrict__ out) {
  long i = (long)blockIdx.x * blockDim.x + threadIdx.x;
  const long total = (long)BB * NN * 513;
  if (i >= total) return;
  const int m = (int)(i % 513);
  const long bn = i / 513;
  const float f = (float)m * (16000.f / 1024.f);
  int sgi; float wt;
  if (f <= 1000.f)      { sgi = 0; wt = f * (1.f / 1000.f); }
  else if (f <= 2000.f) { sgi = 1; wt = (f - 1000.f) * (1.f / 1000.f); }
  else if (f <= 4000.f) { sgi = 2; wt = (f - 2000.f) * (1.f / 2000.f); }
  else                  { sgi = 3; wt = (f - 4000.f) * (1.f / 4000.f); }
  // ap channel c -> band index: c<=1 -> band3 ; else band (4-c)
  const int b0 = (sgi <= 1) ? 3 : (4 - sgi);
  const int c1 = sgi + 1;
  const int b1 = (c1 <= 1) ? 3 : (4 - c1);
  const float y0 = logf(Ab[bn * 4 + b0]);
  const float y1 = logf(Ab[bn * 4 + b1]);
  out[i] = expf(y0 + wt * (y1 - y0));
}

extern "C" void kernel_launch(void* const* d_in, const int* in_sizes, int n_in,
                              void* d_out, int out_size, void* d_ws, size_t ws_size,
                              hipStream_t stream) {
  (void)in_sizes; (void)n_in; (void)out_size; (void)ws_size;
  const float* x  = (const float*)d_in[0];
  const float* f0 = (const float*)d_in[1];
  float* out = (float*)d_out;

  float* ws  = (float*)d_ws;
  float* xb0 = ws;                 // 8*160000
  float* lx1 = xb0 + 1280000;      // 8*160000
  float* xb1 = lx1 + 1280000;      // 8*80000
  float* lx2 = xb1 + 640000;       // 8*80000
  float* xb2 = lx2 + 640000;       // 8*40000
  float* lx3 = xb2 + 320000;       // 8*40000 (= band 3 signal)
  float* Ab  = lx3 + 320000;       // 8*4000*4

  // QMF cascade
  qmf_kernel<<<(BB * 160000 + 255) / 256, 256, 0, stream>>>(x,   xb0, lx1, 320000);
  qmf_kernel<<<(BB *  80000 + 255) / 256, 256, 0, stream>>>(lx1, xb1, lx2, 160000);
  qmf_kernel<<<(BB *  40000 + 255) / 256, 256, 0, stream>>>(lx2, xb2, lx3,  80000);

  // Band aperiodicity: 2 frames per wave, 8 waves per block -> exact grid
  const int nBlocks = (BB * (NN / 2)) / 8;  // 2000
  band_kernel<<<nBlocks, 256, 0, stream>>>(xb0, 160000, f0, Ab, 0, 8000.f, 241);
  band_kernel<<<nBlocks, 256, 0, stream>>>(xb1,  80000, f0, Ab, 1, 4000.f, 121);
  band_kernel<<<nBlocks, 256, 0, stream>>>(xb2,  40000, f0, Ab, 2, 2000.f,  61);
  band_kernel<<<nBlocks, 256, 0, stream>>>(lx3,  40000, f0, Ab, 3, 2000.f,  61);

  // Final spectral interpolation
  const long total = (long)BB * NN * 513;
  interp_kernel<<<(int)((total + 255) / 256), 256, 0, stream>>>(Ab, out);
}